// BLoraQKVColumnParallelLinear_59931973648401
// MI455X (gfx1250) — compile-verified
//
#include <hip/hip_runtime.h>
#include <hip/hip_bf16.h>
#include <stdint.h>

typedef __attribute__((ext_vector_type(2))) float v2f;
typedef __attribute__((ext_vector_type(8))) float v8f;

#define B_   4
#define S_   2048
#define D_   4096
#define OUT_ 12288
#define NA_  8
#define R_   16

// ---------------------------------------------------------------------------
// Kernel 1: LoRA down-projection for the assigned adapter only.
//   t[p][b][s][r] = scaling[a(b)] * sum_d x[b,s,d] * lora_A[p, a(b), r, d]
// ---------------------------------------------------------------------------
__global__ __launch_bounds__(256)
void lora_down_kernel(const float* __restrict__ x,
                      const float* __restrict__ loraA,
                      const float* __restrict__ scaling,
                      const float* __restrict__ masks,
                      float* __restrict__ t)
{
    const int tid = threadIdx.x;
    const int s   = blockIdx.x * 256 + tid;
    const int b   = blockIdx.y;
    const int p   = blockIdx.z;

    int a = 0;
    #pragma unroll
    for (int n = 0; n < NA_; ++n)
        if (masks[n * B_ + b] > 0.5f) a = n;
    const float sc = scaling[a];

    __shared__ float As[R_][128];

    float acc[R_];
    #pragma unroll
    for (int r = 0; r < R_; ++r) acc[r] = 0.0f;

    const float* xrow  = x + ((size_t)b * S_ + s) * D_;
    const float* Abase = loraA + ((size_t)(p * NA_ + a) * R_) * D_;

    for (int kb = 0; kb < D_ / 128; ++kb) {
        __syncthreads();
        #pragma unroll
        for (int q = 0; q < 8; ++q) {
            int idx = tid + q * 256;
            int row = idx >> 7;
            int kk  = idx & 127;
            As[row][kk] = Abase[(size_t)row * D_ + kb * 128 + kk];
        }
        __syncthreads();

        const float4* xv4 = (const float4*)(xrow + kb * 128);
        #pragma unroll 4
        for (int k4 = 0; k4 < 32; ++k4) {
            float4 xv = xv4[k4];
            int k = k4 * 4;
            #pragma unroll
            for (int r = 0; r < R_; ++r) {
                acc[r] += xv.x * As[r][k + 0];
                acc[r] += xv.y * As[r][k + 1];
                acc[r] += xv.z * As[r][k + 2];
                acc[r] += xv.w * As[r][k + 3];
            }
        }
    }

    float* to = t + (((size_t)(p * B_ + b) * S_) + s) * R_;
    #pragma unroll
    for (int r = 0; r < R_; ++r) to[r] = sc * acc[r];
}

// ---------------------------------------------------------------------------
// Kernel 2: fused-QKV GEMM, fp32 WMMA 16x16x4, async global->LDS double buffer
// Block tile 128(M) x 64(N), K-tile 32. 8 waves, each wave owns 16x64.
// ---------------------------------------------------------------------------
#define BM 128
#define BN 64
#define BK 32
#define KPAD 4
#define KST (BK + KPAD)     // 36-dword row stride: conflict-free, rows 16B-aligned
#define NK  (D_ / BK)       // 128 k-tiles

__device__ __forceinline__ void async_cp16(unsigned lds_off, unsigned long long gaddr)
{
    // memory -> LDS direct, tracked by ASYNCcnt (no VGPR round trip)
    asm volatile("global_load_async_to_lds_b128 %0, %1, off"
                 :: "v"(lds_off), "v"(gaddr) : "memory");
}

__global__ __launch_bounds__(256)
void qkv_lora_gemm_kernel(const float* __restrict__ x,
                          const float* __restrict__ w,
                          const float* __restrict__ bias,
                          const float* __restrict__ loraB,
                          const float* __restrict__ masks,
                          const float* __restrict__ t,
                          float* __restrict__ out)
{
    __shared__ __align__(16) float Xs[2][BM][KST];  // 2 x 18 KB
    __shared__ __align__(16) float Ws[2][BN][KST];  // 2 x  9 KB
    __shared__ __align__(16) float Ts[BM][R_];      //      8 KB

    const int tid  = threadIdx.x;
    const int wave = tid >> 5;
    const int lane = tid & 31;
    const int half = lane >> 4;
    const int lr   = lane & 15;

    const int n0 = blockIdx.x * BN;
    const int m0 = blockIdx.y * BM;

    const int b       = m0 / S_;
    const int sBase   = m0 % S_;
    const int p       = n0 / D_;
    const int colBase = n0 % D_;

    int a = 0;
    #pragma unroll
    for (int n = 0; n < NA_; ++n)
        if (masks[n * B_ + b] > 0.5f) a = n;

    // per-thread staging addresses (6 async b128 per k-tile per thread)
    const int c4 = (tid & 7) * 4;
    const int rL = tid >> 3;
    unsigned ldsX[2][4], ldsW[2][2];
    unsigned long long gX[4], gW[2];
    #pragma unroll
    for (int j = 0; j < 4; ++j) {
        int row = rL + 32 * j;
        gX[j] = (unsigned long long)(uintptr_t)(x + (size_t)(m0 + row) * D_ + c4);
        ldsX[0][j] = (unsigned)(uintptr_t)&Xs[0][row][c4];
        ldsX[1][j] = (unsigned)(uintptr_t)&Xs[1][row][c4];
    }
    #pragma unroll
    for (int j = 0; j < 2; ++j) {
        int row = rL + 32 * j;
        gW[j] = (unsigned long long)(uintptr_t)(w + (size_t)(n0 + row) * D_ + c4);
        ldsW[0][j] = (unsigned)(uintptr_t)&Ws[0][row][c4];
        ldsW[1][j] = (unsigned)(uintptr_t)&Ws[1][row][c4];
    }

    // stage the 128 t-rows this block needs (p,b fixed)
    const float* tsrc = t + (((size_t)(p * B_ + b) * S_) + sBase) * R_;
    #pragma unroll
    for (int q = 0; q < 8; ++q) {
        int idx = tid + q * 256;
        ((float*)Ts)[idx] = tsrc[idx];
    }

    v8f acc[4];
    #pragma unroll
    for (int j = 0; j < 4; ++j) acc[j] = (v8f)0.0f;

    // prologue: k-tile 0 -> buffer 0
    #pragma unroll
    for (int j = 0; j < 4; ++j) async_cp16(ldsX[0][j], gX[j]);
    #pragma unroll
    for (int j = 0; j < 2; ++j) async_cp16(ldsW[0][j], gW[j]);

    for (int kb = 0; kb < NK; ++kb) {
        const int cur = kb & 1;
        if (kb + 1 < NK) {
            const unsigned long long koff = (unsigned long long)(kb + 1) * (BK * 4);
            const int nxt = cur ^ 1;
            #pragma unroll
            for (int j = 0; j < 4; ++j) async_cp16(ldsX[nxt][j], gX[j] + koff);
            #pragma unroll
            for (int j = 0; j < 2; ++j) async_cp16(ldsW[nxt][j], gW[j] + koff);
            // in-order completion: <=6 outstanding => current tile landed
            asm volatile("s_wait_asynccnt 0x6" ::: "memory");
        } else {
            asm volatile("s_wait_asynccnt 0x0" ::: "memory");
        }
        __syncthreads();

        #pragma unroll
        for (int kk = 0; kk < BK / 4; ++kk) {
            const int k = kk * 4 + half * 2;   // A/B frag: VGPR0 K={0,2}, VGPR1 K={1,3}
            v2f av;
            av.x = Xs[cur][wave * 16 + lr][k];
            av.y = Xs[cur][wave * 16 + lr][k + 1];
            #pragma unroll
            for (int j = 0; j < 4; ++j) {
                v2f bv;
                bv.x = Ws[cur][j * 16 + lr][k];
                bv.y = Ws[cur][j * 16 + lr][k + 1];
                acc[j] = __builtin_amdgcn_wmma_f32_16x16x4_f32(
                    false, av, false, bv, (short)0, acc[j], false, false);
            }
        }
        __syncthreads();   // all waves done reading buf[cur] before it is overwritten
    }

    // Epilogue: bias + rank-16 LoRA up-projection fused into the store.
    #pragma unroll
    for (int j = 0; j < 4; ++j) {
        const int nLoc  = j * 16 + lr;
        const int nGlob = n0 + nLoc;
        const float bi  = bias[nGlob];
        const int col   = colBase + nLoc;
        const float4* bp = (const float4*)(loraB +
            (((size_t)(p * NA_ + a) * D_) + col) * R_);
        const float4 lb0 = bp[0], lb1 = bp[1], lb2 = bp[2], lb3 = bp[3];
        #pragma unroll
        for (int i = 0; i < 8; ++i) {
            const int rowL = wave * 16 + i + 8 * half;   // C/D layout: VGPRi -> M=i / i+8
            const float4* tr = (const float4*)(&Ts[rowL][0]);
            const float4 t0 = tr[0], t1 = tr[1], t2 = tr[2], t3 = tr[3];
            float lc = t0.x*lb0.x + t0.y*lb0.y + t0.z*lb0.z + t0.w*lb0.w
                     + t1.x*lb1.x + t1.y*lb1.y + t1.z*lb1.z + t1.w*lb1.w
                     + t2.x*lb2.x + t2.y*lb2.y + t2.z*lb2.z + t2.w*lb2.w
                     + t3.x*lb3.x + t3.y*lb3.y + t3.z*lb3.z + t3.w*lb3.w;
            out[(size_t)(m0 + rowL) * OUT_ + nGlob] = acc[j][i] + bi + lc;
        }
    }
}

// ---------------------------------------------------------------------------
extern "C" void kernel_launch(void* const* d_in, const int* in_sizes, int n_in,
                              void* d_out, int out_size, void* d_ws, size_t ws_size,
                              hipStream_t stream)
{
    const float* x       = (const float*)d_in[0];
    const float* weight  = (const float*)d_in[1];
    const float* bias    = (const float*)d_in[2];
    const float* loraA   = (const float*)d_in[3];
    const float* loraB   = (const float*)d_in[4];
    const float* scaling = (const float*)d_in[5];
    const float* masks   = (const float*)d_in[6];
    float* out = (float*)d_out;
    float* t   = (float*)d_ws;        // 3*B*S*R floats = 1.5 MB scratch

    dim3 g1(S_ / 256, B_, 3);
    lora_down_kernel<<<g1, 256, 0, stream>>>(x, loraA, scaling, masks, t);

    dim3 g2(OUT_ / BN, (B_ * S_) / BM);
    qkv_lora_gemm_kernel<<<g2, 256, 0, stream>>>(x, weight, bias, loraB,
                                                 masks, t, out);
}